// SageGCN_1735166787609
// MI455X (gfx1250) — compile-verified
//
#include <hip/hip_runtime.h>

typedef __attribute__((ext_vector_type(2))) float v2f;
typedef __attribute__((ext_vector_type(8))) float v8f;

#define D 128
#define NODES_PER_BLOCK 32
#define LDS_STRIDE 260   // 256 + 4 pad: conflict-free A-fragment reads (260 % 64 == 4)

// ---------------------------------------------------------------------------
// Kernel 1: exclusive prefix sum of counts (action[i]+1) -> start offsets
// Single block, chunked Hillis-Steele scan with running carry.
// ---------------------------------------------------------------------------
__global__ __launch_bounds__(1024)
void sage_scan_offsets(const int* __restrict__ action, int* __restrict__ start, int n) {
    __shared__ int buf[1024];
    __shared__ int carry_s;
    if (threadIdx.x == 0) carry_s = 0;
    __syncthreads();
    for (int base = 0; base < n; base += 1024) {
        const int i = base + (int)threadIdx.x;
        const int v = (i < n) ? (action[i] + 1) : 0;
        buf[threadIdx.x] = v;
        __syncthreads();
        for (int off = 1; off < 1024; off <<= 1) {
            const int t = (threadIdx.x >= (unsigned)off) ? buf[threadIdx.x - off] : 0;
            __syncthreads();
            buf[threadIdx.x] += t;
            __syncthreads();
        }
        const int incl  = buf[threadIdx.x];
        const int carry = carry_s;
        if (i < n) start[i] = carry + incl - v;   // exclusive offset
        __syncthreads();
        if (threadIdx.x == 1023) carry_s = carry + buf[1023];
        __syncthreads();
    }
}

// ---------------------------------------------------------------------------
// Kernel 2 (fused): per 32-node tile
//   phase 1: segment-mean of neighbor rows + stage src rows into LDS [32 x 256]
//   phase 2: C = [aggr|src] x [weight;b] via V_WMMA_F32_16X16X4_F32, ReLU, store
// ---------------------------------------------------------------------------
__global__ __launch_bounds__(256)
void sage_fused(const int*   __restrict__ action,
                const float* __restrict__ src,
                const float* __restrict__ nbr,
                const float* __restrict__ weight,
                const float* __restrict__ bmat,
                const int*   __restrict__ start,
                float*       __restrict__ out, int n)
{
    __shared__ float tile[NODES_PER_BLOCK * LDS_STRIDE];
    const int tid   = (int)threadIdx.x;
    const int wave  = tid >> 5;
    const int lane  = tid & 31;
    const int node0 = (int)blockIdx.x * NODES_PER_BLOCK;

    // ---- Phase 1: segment mean + src staging (each wave owns 4 nodes) ----
    for (int i = 0; i < 4; ++i) {
        const int r    = wave * 4 + i;        // 0..31 tile row
        const int node = node0 + r;
        float* row = &tile[r * LDS_STRIDE];
        float4 acc = make_float4(0.f, 0.f, 0.f, 0.f);
        float4 s   = make_float4(0.f, 0.f, 0.f, 0.f);
        if (node < n) {
            const int cnt = action[node] + 1;  // >= 1
            const int off = start[node];
            for (int j = 0; j < cnt; ++j) {    // 512B coalesced per row
                const float4 v = *(const float4*)&nbr[((size_t)(off + j)) * D + lane * 4];
                acc.x += v.x; acc.y += v.y; acc.z += v.z; acc.w += v.w;
            }
            const float inv = 1.0f / (float)cnt;
            acc.x *= inv; acc.y *= inv; acc.z *= inv; acc.w *= inv;
            s = *(const float4*)&src[(size_t)node * D + lane * 4];
        }
        *(float4*)&row[lane * 4]     = acc;    // cols   0..127 : aggr
        *(float4*)&row[D + lane * 4] = s;      // cols 128..255 : src
    }
    __syncthreads();

    // ---- Phase 2: WMMA GEMM over concatenated K = 256 ----
    const int mt   = (wave >> 2) * 16;         // M tile base (0 or 16)
    const int nt0  = (wave & 3) * 16;          // first N tile
    const int nt1  = nt0 + 64;                 // second N tile
    const int koff = (lane < 16) ? 0 : 2;      // K split across half-waves
    const int bn   = lane & 15;
    const float* arowp = &tile[(mt + bn) * LDS_STRIDE + koff];

    v8f c0 = {}; v8f c1 = {};

    #pragma unroll 4
    for (int k = 0; k < D; k += 4) {           // aggr @ weight
        v2f a; a.x = arowp[k]; a.y = arowp[k + 1];
        const float* bp = &weight[(k + koff) * D + bn];
        v2f b0; b0.x = bp[nt0]; b0.y = bp[D + nt0];
        v2f b1; b1.x = bp[nt1]; b1.y = bp[D + nt1];
        c0 = __builtin_amdgcn_wmma_f32_16x16x4_f32(false, a, false, b0, (short)0, c0, false, false);
        c1 = __builtin_amdgcn_wmma_f32_16x16x4_f32(false, a, false, b1, (short)0, c1, false, false);
    }
    #pragma unroll 4
    for (int k = 0; k < D; k += 4) {           // src @ b, accumulated into same C
        v2f a; a.x = arowp[D + k]; a.y = arowp[D + k + 1];
        const float* bp = &bmat[(k + koff) * D + bn];
        v2f b0; b0.x = bp[nt0]; b0.y = bp[D + nt0];
        v2f b1; b1.x = bp[nt1]; b1.y = bp[D + nt1];
        c0 = __builtin_amdgcn_wmma_f32_16x16x4_f32(false, a, false, b0, (short)0, c0, false, false);
        c1 = __builtin_amdgcn_wmma_f32_16x16x4_f32(false, a, false, b1, (short)0, c1, false, false);
    }

    // ---- ReLU + store (C/D layout: M = j + (lane<16?0:8), N = lane&15) ----
    const int mbase = mt + ((lane < 16) ? 0 : 8);
    #pragma unroll
    for (int j = 0; j < 8; ++j) {
        const int row = node0 + mbase + j;
        if (row < n) {
            out[(size_t)row * D + nt0 + bn] = fmaxf(c0[j], 0.0f);
            out[(size_t)row * D + nt1 + bn] = fmaxf(c1[j], 0.0f);
        }
    }
}

// ---------------------------------------------------------------------------
extern "C" void kernel_launch(void* const* d_in, const int* in_sizes, int n_in,
                              void* d_out, int out_size, void* d_ws, size_t ws_size,
                              hipStream_t stream) {
    const int*   action = (const int*)d_in[0];
    const float* src    = (const float*)d_in[1];
    const float* nbr    = (const float*)d_in[2];
    const float* weight = (const float*)d_in[3];
    const float* bmat   = (const float*)d_in[4];
    float* out = (float*)d_out;
    const int n = in_sizes[0];

    int* start = (int*)d_ws;   // n ints of scratch (~200 KB)

    sage_scan_offsets<<<1, 1024, 0, stream>>>(action, start, n);

    const int blocks = (n + NODES_PER_BLOCK - 1) / NODES_PER_BLOCK;
    sage_fused<<<blocks, 256, 0, stream>>>(action, src, nbr, weight, bmat, start, out, n);
}